// CANLayer_78134045048954
// MI455X (gfx1250) — compile-verified
//
#include <hip/hip_runtime.h>
#include <hip/hip_bf16.h>

typedef __attribute__((ext_vector_type(2))) float v2f;
typedef __attribute__((ext_vector_type(8))) float v8f;

#define IN_CH 256
#define HC    128          // HEADS * OUT_CH
#define CAN_EPS 1.000001f  // 1.0 + 1e-6
#define SLOPE 0.01f

__device__ __forceinline__ float lrelu(float v) {
    return v > 0.0f ? v : v * SLOPE;
}

// -----------------------------------------------------------------------------
// Fused 3-way GEMM: xm_l = x@Wl, xm_u = x@Wu, acc = EPS*(x@Ws)
// One 16x16 output tile per wave via V_WMMA_F32_16X16X4_F32 (exact fp32).
// A tile (16 rows x 256 K) staged in LDS, shared by the block's 4 waves.
// grid = ( ceil(N/16), 6 ), block = 128 (4 waves); 24 column tiles total.
// -----------------------------------------------------------------------------
__global__ __launch_bounds__(128) void can_gemm_kernel(
    const float* __restrict__ x,
    const float* __restrict__ Wl, const float* __restrict__ Wu,
    const float* __restrict__ Ws,
    float* __restrict__ xm_l, float* __restrict__ xm_u,
    float* __restrict__ acc_out, int N)
{
    __shared__ float lds[16 * IN_CH];   // 16 KB

    const int row0 = blockIdx.x * 16;
    if (row0 >= N) return;
    const int rows = min(16, N - row0);

    // Cooperative load: rows*256 floats, contiguous in memory.
    {
        const float4* s4 = (const float4*)(x + (size_t)row0 * IN_CH);
        float4* d4 = (float4*)lds;
        const int n4 = rows * (IN_CH / 4);
        for (int i = threadIdx.x; i < n4; i += 128) d4[i] = s4[i];
        for (int i = n4 + (int)threadIdx.x; i < 16 * (IN_CH / 4); i += 128)
            d4[i] = make_float4(0.f, 0.f, 0.f, 0.f);
    }
    __syncthreads();

    const int wave = threadIdx.x >> 5;
    const int lane = threadIdx.x & 31;
    const int ct   = blockIdx.y * 4 + wave;   // [0,24): 8 lower | 8 upper | 8 skip

    const float* W; float* dst; int wc; float scale = 1.0f;
    if (ct < 8)       { W = Wl; dst = xm_l;    wc = ct * 16; }
    else if (ct < 16) { W = Wu; dst = xm_u;    wc = (ct - 8) * 16; }
    else              { W = Ws; dst = acc_out; wc = (ct - 16) * 16; scale = CAN_EPS; }

    // WMMA f32 16x16x4 per-lane fragment layout:
    //   A (16x4): lanes 0-15 -> M=lane, K={0,1}; lanes 16-31 -> M=lane-16, K={2,3}
    //   B (4x16): lanes 0-15 -> N=lane, K={0,1}; lanes 16-31 -> N=lane-16, K={2,3}
    const int m  = lane & 15;
    const int kb = (lane >> 4) << 1;            // 0 or 2
    const float* arow = lds + m * IN_CH;
    const float* bcol = W + wc + (lane & 15);   // column of W, stride HC per K

    v8f c = {};
    #pragma unroll 4
    for (int k = 0; k < IN_CH; k += 4) {
        v2f a, b;
        a[0] = arow[k + kb];
        a[1] = arow[k + kb + 1];
        b[0] = bcol[(size_t)(k + kb) * HC];
        b[1] = bcol[(size_t)(k + kb + 1) * HC];
        c = __builtin_amdgcn_wmma_f32_16x16x4_f32(
                /*neg_a=*/false, a, /*neg_b=*/false, b,
                /*c_mod=*/(short)0, c, /*reuse_a=*/false, /*reuse_b=*/false);
    }

    // C/D layout: VGPR r -> lanes 0-15: M=r, N=lane ; lanes 16-31: M=r+8, N=lane-16
    const int nn    = lane & 15;
    const int mbase = (lane >> 4) << 3;
    float* base = dst + (size_t)(row0 + mbase) * HC + wc + nn;
    if (row0 + 16 <= N) {
        // Full tile (always true for N % 16 == 0): branch-free store clause.
        #pragma unroll
        for (int r = 0; r < 8; ++r)
            base[(size_t)r * HC] = c[r] * scale;
    } else {
        #pragma unroll
        for (int r = 0; r < 8; ++r)
            if (row0 + mbase + r < N) base[(size_t)r * HC] = c[r] * scale;
    }
}

// -----------------------------------------------------------------------------
// Per-node attention scalars: a_s[n][h] = leaky_relu(sum_c xm[n][h*64+c]*a_src[h*64+c])
// One wave per node; 8 dot products (lower/upper x src/dst x 2 heads).
// -----------------------------------------------------------------------------
__global__ __launch_bounds__(256) void can_alpha_kernel(
    const float* __restrict__ xm_l, const float* __restrict__ xm_u,
    const float* __restrict__ a_src_l, const float* __restrict__ a_dst_l,
    const float* __restrict__ a_src_u, const float* __restrict__ a_dst_u,
    float* __restrict__ as_l, float* __restrict__ at_l,
    float* __restrict__ as_u, float* __restrict__ at_u, int N)
{
    const int n = blockIdx.x * 8 + (threadIdx.x >> 5);
    if (n >= N) return;
    const int lane = threadIdx.x & 31;

    const float* xl = xm_l + (size_t)n * HC;
    const float* xu = xm_u + (size_t)n * HC;
    const float xl0 = xl[lane], xl1 = xl[lane + 32], xl2 = xl[lane + 64], xl3 = xl[lane + 96];
    const float xu0 = xu[lane], xu1 = xu[lane + 32], xu2 = xu[lane + 64], xu3 = xu[lane + 96];

    float p[8];
    p[0] = xl0 * a_src_l[lane]      + xl1 * a_src_l[lane + 32];   // lower src h0
    p[1] = xl2 * a_src_l[lane + 64] + xl3 * a_src_l[lane + 96];   // lower src h1
    p[2] = xl0 * a_dst_l[lane]      + xl1 * a_dst_l[lane + 32];   // lower dst h0
    p[3] = xl2 * a_dst_l[lane + 64] + xl3 * a_dst_l[lane + 96];   // lower dst h1
    p[4] = xu0 * a_src_u[lane]      + xu1 * a_src_u[lane + 32];
    p[5] = xu2 * a_src_u[lane + 64] + xu3 * a_src_u[lane + 96];
    p[6] = xu0 * a_dst_u[lane]      + xu1 * a_dst_u[lane + 32];
    p[7] = xu2 * a_dst_u[lane + 64] + xu3 * a_dst_u[lane + 96];

    #pragma unroll
    for (int off = 16; off; off >>= 1) {
        #pragma unroll
        for (int i = 0; i < 8; ++i) p[i] += __shfl_xor(p[i], off, 32);
    }

    if (lane == 0) {
        as_l[n * 2 + 0] = lrelu(p[0]);  as_l[n * 2 + 1] = lrelu(p[1]);
        at_l[n * 2 + 0] = lrelu(p[2]);  at_l[n * 2 + 1] = lrelu(p[3]);
        as_u[n * 2 + 0] = lrelu(p[4]);  as_u[n * 2 + 1] = lrelu(p[5]);
        at_u[n * 2 + 0] = lrelu(p[6]);  at_u[n * 2 + 1] = lrelu(p[7]);
    }
}

// -----------------------------------------------------------------------------
// Edge pass: one wave per edge. 2-head softmax, then atomic scatter-add of the
// 128-float weighted message into acc[tgt]. All node arrays are L2-resident.
// -----------------------------------------------------------------------------
__global__ __launch_bounds__(256) void can_edge_kernel(
    const int* __restrict__ src, const int* __restrict__ tgt,
    const float* __restrict__ vals, const float* __restrict__ xm,
    const float* __restrict__ a_s, const float* __restrict__ a_t,
    float* __restrict__ acc, int E)
{
    const int e = blockIdx.x * 8 + (threadIdx.x >> 5);
    if (e >= E) return;
    const int lane = threadIdx.x & 31;

    const int s = src[e], t = tgt[e];
    const float v  = vals[e];
    const float s0 = v + a_s[s * 2 + 0] + a_t[t * 2 + 0];
    const float s1 = v + a_s[s * 2 + 1] + a_t[t * 2 + 1];
    const float mx = fmaxf(s0, s1);
    const float e0 = __expf(s0 - mx);
    const float e1 = __expf(s1 - mx);
    const float inv = 1.0f / (e0 + e1);
    const float w0 = e0 * inv, w1 = e1 * inv;

    const float* xr = xm + (size_t)s * HC;
    float* ar = acc + (size_t)t * HC;
    atomicAdd(&ar[lane],      xr[lane]      * w0);   // head 0: c in [0,64)
    atomicAdd(&ar[lane + 32], xr[lane + 32] * w0);
    atomicAdd(&ar[lane + 64], xr[lane + 64] * w1);   // head 1: c in [64,128)
    atomicAdd(&ar[lane + 96], xr[lane + 96] * w1);
}

// -----------------------------------------------------------------------------
// Final ReLU, in place on d_out, float4-vectorized.
// -----------------------------------------------------------------------------
__global__ __launch_bounds__(256) void can_relu_kernel(float* __restrict__ out, int n4)
{
    const int i = blockIdx.x * 256 + threadIdx.x;
    if (i >= n4) return;
    float4* p = (float4*)out;
    float4 v = p[i];
    v.x = fmaxf(v.x, 0.f); v.y = fmaxf(v.y, 0.f);
    v.z = fmaxf(v.z, 0.f); v.w = fmaxf(v.w, 0.f);
    p[i] = v;
}

extern "C" void kernel_launch(void* const* d_in, const int* in_sizes, int n_in,
                              void* d_out, int out_size, void* d_ws, size_t ws_size,
                              hipStream_t stream) {
    const float* x      = (const float*)d_in[0];
    const int*   l_tgt  = (const int*)  d_in[1];
    const int*   l_src  = (const int*)  d_in[2];
    const float* l_vals = (const float*)d_in[3];
    const int*   u_tgt  = (const int*)  d_in[4];
    const int*   u_src  = (const int*)  d_in[5];
    const float* u_vals = (const float*)d_in[6];
    const float* Wl     = (const float*)d_in[7];
    const float* asl_w  = (const float*)d_in[8];
    const float* adl_w  = (const float*)d_in[9];
    const float* Wu     = (const float*)d_in[10];
    const float* asu_w  = (const float*)d_in[11];
    const float* adu_w  = (const float*)d_in[12];
    const float* Ws     = (const float*)d_in[13];

    const int N = in_sizes[0] / IN_CH;   // 50000
    const int E = in_sizes[1];           // 800000
    float* out = (float*)d_out;          // doubles as accumulator (init = EPS*skip)

    // Workspace carve-up (floats): xm_l | xm_u | 4 alpha arrays of 2N each.
    float* ws   = (float*)d_ws;
    float* xm_l = ws;
    float* xm_u = xm_l + (size_t)N * HC;
    float* as_l = xm_u + (size_t)N * HC;
    float* at_l = as_l + 2 * (size_t)N;
    float* as_u = at_l + 2 * (size_t)N;
    float* at_u = as_u + 2 * (size_t)N;

    // 1) fused GEMM: 24 col-tiles (8 per matrix), 4 waves/block -> grid.y = 6
    dim3 ggrid((N + 15) / 16, 6);
    can_gemm_kernel<<<ggrid, 128, 0, stream>>>(x, Wl, Wu, Ws, xm_l, xm_u, out, N);

    // 2) per-node attention scalars (one wave per node, 8 waves/block)
    can_alpha_kernel<<<(N + 7) / 8, 256, 0, stream>>>(
        xm_l, xm_u, asl_w, adl_w, asu_w, adu_w, as_l, at_l, as_u, at_u, N);

    // 3) edge scatter passes (one wave per edge)
    can_edge_kernel<<<(E + 7) / 8, 256, 0, stream>>>(l_src, l_tgt, l_vals, xm_l, as_l, at_l, out, E);
    can_edge_kernel<<<(E + 7) / 8, 256, 0, stream>>>(u_src, u_tgt, u_vals, xm_u, as_u, at_u, out, E);

    // 4) final ReLU in place
    const int n4 = (N * HC) / 4;
    can_relu_kernel<<<(n4 + 255) / 256, 256, 0, stream>>>(out, n4);
}